// LinearAttention3D_70944269795782
// MI455X (gfx1250) — compile-verified
//
#include <hip/hip_runtime.h>
#include <hip/hip_bf16.h>

// LinearAttention3D for MI455X (gfx1250), wave32, f32 WMMA 16x16x4,
// async global->LDS staging (ASYNCcnt path).
//
// Shapes: B=2, DIM=128, HEADS=4, DIM_HEAD=2, HIDDEN=8, n = 128*128*16 = 262144.
// Inputs (f32): x, query, W_qkv[24,128], W_q[8,128], W_out[128,8], b_out[128],
//               gamma[128], beta[128].  Output f32 [2,128,n].
//
// Workspace layout (floats):
//   qws  = ws[0 .. NBVOX*8)                  : softmaxed q per voxel (16 MiB)
//   acc  = ws + NBVOX*8 (small region):
//     [0..15]   sumexp[b][8]        (Σ_n exp(k))
//     [16..47]  ctx_un[b][h][d][e]  (Σ_n exp(k_d) v_e)
//     [48..63]  sumq[b][8]          (Σ_n q_soft)
//     [64..65]  s2[b]               (Σ_n aᵀGa + 2h·a)
//     [66..67]  mean[b]   [68..69] rstd[b]
//     [72..199] T[b][8][8]          (a = T q)
//     [200..327] Q8[b][8][8] = TᵀGT
//     [328..343] L8[b][8]    = Tᵀh
//     [344..351] u[8] = Σ_c W_out[c,:]
//     [352] Σb   [353] Σb²

#define DIMC   128
#define NVOX   (128 * 128 * 16)   // 262144 = 2^18
#define NBVOX  (2 * NVOX)
#define LOG2N  18
#define TSTRIDE 72                // tile channel stride (bank-conflict free)

typedef float v2f __attribute__((ext_vector_type(2)));
typedef float v8f __attribute__((ext_vector_type(8)));
typedef int   v4i __attribute__((ext_vector_type(4)));

#if __has_builtin(__builtin_amdgcn_global_load_async_to_lds_b128) && \
    __has_builtin(__builtin_amdgcn_s_wait_asynccnt)
#define HAVE_ASYNC 1
#else
#define HAVE_ASYNC 0
#endif

#define GLOBAL_AS __attribute__((address_space(1)))
#define LDS_AS    __attribute__((address_space(3)))

__device__ __forceinline__ v8f wmma4(v2f a, v2f b, v8f c) {
  // V_WMMA_F32_16X16X4_F32 : D(16x16,f32) = A(16x4) * B(4x16) + C
  return __builtin_amdgcn_wmma_f32_16x16x4_f32(false, a, false, b, (short)0, c,
                                               false, false);
}

// Stage a 128ch x 64vox f32 tile into LDS (row-major, stride TSTRIDE floats).
// Async path: GLOBAL_LOAD_ASYNC_TO_LDS_B128, no VGPR round trip, ASYNCcnt.
__device__ __forceinline__ void stage_tile(const float* __restrict__ src,
                                           float* __restrict__ tileL, int tid) {
#if HAVE_ASYNC
  for (int i = tid; i < 2048; i += 128) {
    int c = i >> 4, q4 = (i & 15) << 2;
    __builtin_amdgcn_global_load_async_to_lds_b128(
        (GLOBAL_AS v4i*)(src + (size_t)c * NVOX + q4),
        (LDS_AS v4i*)(tileL + c * TSTRIDE + q4), 0, 0);
  }
  __builtin_amdgcn_s_wait_asynccnt(0);
#else
  for (int i = tid; i < 2048; i += 128) {
    int c = i >> 4, q4 = (i & 15) << 2;
    float4 t = *(const float4*)(src + (size_t)c * NVOX + q4);
    float* d = tileL + c * TSTRIDE + q4;
    d[0] = t.x; d[1] = t.y; d[2] = t.z; d[3] = t.w;
  }
#endif
}

// ---------------------------------------------------------------------------
// Kernel A: per 64-voxel macro-tile (4 waves x 16 voxels each):
//   KV16 = W_kv(16x128) @ X(128x16)     -> exp(k), context partials, sumexp
//   Q16  = W_qp(16x128) @ Query(128x16) -> per-head softmax * scale, store, Σq
// Weights packed in LDS as (k,k+1) pairs (one ds b64 per WMMA A operand,
// matching f32 A layout: K0/K1 low lanes, K2/K3 high lanes). Data tile is
// row-major stride-72: B operand pair = ds_load_2addr_b32 offset1:72.
// ---------------------------------------------------------------------------
__global__ __launch_bounds__(128) void ka_project(
    const float* __restrict__ x, const float* __restrict__ query,
    const float* __restrict__ Wqkv, const float* __restrict__ Wq,
    float* __restrict__ qws, float* __restrict__ acc)
{
  __shared__ __align__(16) float wkvL[2048];
  __shared__ __align__(16) float wqL[2048];
  __shared__ __align__(16) float tileL[128 * TSTRIDE];
  __shared__ float accL[64];

  const int tid  = threadIdx.x;
  const int lane = tid & 31;
  const int wv   = tid >> 5;     // wave in block, 0..3
  const int m    = lane & 15;
  const int half = lane >> 4;

  // pack weights: P(k,m) = (k>>1)*32 + m*2 + (k&1)
  for (int i = tid; i < 2048; i += 128) {
    int mm = i & 15, k = i >> 4;
    int p = ((k >> 1) << 5) + (mm << 1) + (k & 1);
    wkvL[p] = Wqkv[(8 + mm) * DIMC + k];              // rows 8..23: k then v
    wqL[p]  = (mm < 8) ? Wq[mm * DIMC + k] : 0.0f;    // pad to 16 rows
  }
  if (tid < 64) accL[tid] = 0.0f;
  __syncthreads();

  const int j0 = wv << 4;                   // this wave's voxel offset in tile
  const float scale = 0.7071067811865475f;  // DIM_HEAD^-0.5

  for (int mt = blockIdx.x; mt < (NBVOX / 64); mt += gridDim.x) {
    const int vbase = mt << 6;
    const int b  = vbase >> LOG2N;
    const int n0 = vbase & (NVOX - 1);
    const float* xg = x     + (size_t)b * DIMC * NVOX + n0;
    const float* qg = query + (size_t)b * DIMC * NVOX + n0;

    // ---- stage x tile ----
    stage_tile(xg, tileL, tid);
    __syncthreads();

    // ---- KV chain: 32 x wmma_f32_16x16x4 over K=128 ----
    v8f akv = {};
    {
      int oa = (half << 5) + (m << 1);
      int ob = (half * 2) * TSTRIDE + j0 + m;   // k0 = 2*half
      #pragma unroll
      for (int s = 0; s < 32; ++s) {
        v2f a  = *(const v2f*)&wkvL[oa];
        v2f bb = { tileL[ob], tileL[ob + TSTRIDE] };
        akv = wmma4(a, bb, akv);
        oa += 64; ob += 4 * TSTRIDE;
      }
    }
    // D layout: vgpr j -> channel j (low lanes: k) / j+8 (high lanes: v),
    // column = lane&15 = voxel. Swap halves to pair k with v per voxel.
    float ek[8], vvv[8];
    #pragma unroll
    for (int j = 0; j < 8; ++j) {
      vvv[j] = __shfl_xor(akv[j], 16);
      ek[j]  = __expf(akv[j]);
    }
    float red[24];
    #pragma unroll
    for (int j = 0; j < 8; ++j) red[j] = ek[j];
    #pragma unroll
    for (int h = 0; h < 4; ++h)
      #pragma unroll
      for (int d = 0; d < 2; ++d)
        #pragma unroll
        for (int e = 0; e < 2; ++e)
          red[8 + h * 4 + d * 2 + e] = ek[h * 2 + d] * vvv[h * 2 + e];
    #pragma unroll
    for (int i = 0; i < 24; ++i) {        // reduce low 16 lanes (valid half)
      float r = red[i];
      r += __shfl_xor(r, 1); r += __shfl_xor(r, 2);
      r += __shfl_xor(r, 4); r += __shfl_xor(r, 8);
      red[i] = r;
    }
    if (lane == 0) {
      #pragma unroll
      for (int j = 0; j < 8; ++j)  atomicAdd(&accL[b * 8 + j], red[j]);
      #pragma unroll
      for (int i = 0; i < 16; ++i) atomicAdd(&accL[16 + b * 16 + i], red[8 + i]);
    }
    __syncthreads();

    // ---- stage query tile (reuse buffer) ----
    stage_tile(qg, tileL, tid);
    __syncthreads();

    // ---- Q chain ----
    v8f aq = {};
    {
      int oa = (half << 5) + (m << 1);
      int ob = (half * 2) * TSTRIDE + j0 + m;
      #pragma unroll
      for (int s = 0; s < 32; ++s) {
        v2f a  = *(const v2f*)&wqL[oa];
        v2f bb = { tileL[ob], tileL[ob + TSTRIDE] };
        aq = wmma4(a, bb, aq);
        oa += 64; ob += 4 * TSTRIDE;
      }
    }
    // softmax over dim_head (pairs) * scale; valid on low 16 lanes
    float qs[8];
    #pragma unroll
    for (int h = 0; h < 4; ++h) {
      float e0 = __expf(aq[2 * h]), e1 = __expf(aq[2 * h + 1]);
      float r = scale / (e0 + e1);
      qs[2 * h] = e0 * r; qs[2 * h + 1] = e1 * r;
    }
    if (lane < 16) {
      float* dst = qws + ((size_t)(vbase + j0 + lane) << 3);
      *(float4*)(dst)     = make_float4(qs[0], qs[1], qs[2], qs[3]);
      *(float4*)(dst + 4) = make_float4(qs[4], qs[5], qs[6], qs[7]);
    }
    #pragma unroll
    for (int j = 0; j < 8; ++j) {
      float r = qs[j];
      r += __shfl_xor(r, 1); r += __shfl_xor(r, 2);
      r += __shfl_xor(r, 4); r += __shfl_xor(r, 8);
      if (lane == 0) atomicAdd(&accL[48 + b * 8 + j], r);
    }
    __syncthreads();
  }
  if (tid < 64) unsafeAtomicAdd(&acc[tid], accL[tid]);
}

// ---------------------------------------------------------------------------
// Kernel B: finalize context -> T; compute G, h, u, Σb, Σb²; Q8=TᵀGT, L8=Tᵀh.
// ---------------------------------------------------------------------------
__global__ __launch_bounds__(256) void kb_finalize(
    const float* __restrict__ Wout, const float* __restrict__ bout,
    float* __restrict__ acc)
{
  __shared__ float Tl[128], Gl[64], hl[8];
  const int tid = threadIdx.x;
  if (tid < 128) {  // T[b][o=(h,e)][i=(h,d)] = ctx_un/sumexp (block-diagonal)
    int b = tid >> 6, o = (tid >> 3) & 7, i = tid & 7;
    int ho = o >> 1, e = o & 1, hi = i >> 1, d = i & 1;
    float v = 0.0f;
    if (ho == hi)
      v = acc[16 + b * 16 + ho * 4 + d * 2 + e] / acc[b * 8 + ho * 2 + d];
    Tl[tid] = v;
    acc[72 + tid] = v;
  } else if (tid < 192) {                 // G[e][f] = Σ_c W[c,e] W[c,f]
    int e = (tid - 128) >> 3, f = (tid - 128) & 7;
    float s = 0.0f;
    for (int c = 0; c < DIMC; ++c) s += Wout[c * 8 + e] * Wout[c * 8 + f];
    Gl[tid - 128] = s;
  } else if (tid < 200) {                 // h[e] = Σ_c W[c,e] b[c]
    int e = tid - 192; float s = 0.0f;
    for (int c = 0; c < DIMC; ++c) s += Wout[c * 8 + e] * bout[c];
    hl[e] = s;
  } else if (tid < 208) {                 // u[e] = Σ_c W[c,e]
    int e = tid - 200; float s = 0.0f;
    for (int c = 0; c < DIMC; ++c) s += Wout[c * 8 + e];
    acc[344 + e] = s;
  } else if (tid == 208) {
    float s = 0.0f; for (int c = 0; c < DIMC; ++c) s += bout[c];
    acc[352] = s;
  } else if (tid == 209) {
    float s = 0.0f; for (int c = 0; c < DIMC; ++c) s += bout[c] * bout[c];
    acc[353] = s;
  }
  __syncthreads();
  if (tid < 128) {                        // Q8[b][i][j] = Σ_ef T[e,i]G[e,f]T[f,j]
    int b = tid >> 6, i = (tid >> 3) & 7, j = tid & 7;
    float s = 0.0f;
    for (int e = 0; e < 8; ++e) {
      float inner = 0.0f;
      for (int f = 0; f < 8; ++f) inner += Gl[e * 8 + f] * Tl[b * 64 + f * 8 + j];
      s += Tl[b * 64 + e * 8 + i] * inner;
    }
    acc[200 + tid] = s;
  } else if (tid < 144) {                 // L8[b][i] = Σ_e T[e,i] h[e]
    int b = (tid - 128) >> 3, i = (tid - 128) & 7;
    float s = 0.0f;
    for (int e = 0; e < 8; ++e) s += Tl[b * 64 + e * 8 + i] * hl[e];
    acc[328 + (tid - 128)] = s;
  }
}

// ---------------------------------------------------------------------------
// Kernel C: s2[b] = Σ_n ( qᵀ Q8 q + 2 L8·q )
// ---------------------------------------------------------------------------
__global__ __launch_bounds__(256) void kc_stats(
    const float* __restrict__ qws, float* __restrict__ acc)
{
  __shared__ float Q8l[128], L8l[16], s2L[2];
  const int tid = threadIdx.x;
  if (tid < 128) Q8l[tid] = acc[200 + tid];
  if (tid < 16)  L8l[tid] = acc[328 + tid];
  if (tid < 2)   s2L[tid] = 0.0f;
  __syncthreads();

  float s2a[2] = {0.0f, 0.0f};
  const int gstride = gridDim.x * blockDim.x;
  for (int vox = blockIdx.x * blockDim.x + tid; vox < NBVOX; vox += gstride) {
    const int b = vox >> LOG2N;
    const float* qp = qws + ((size_t)vox << 3);
    float4 p0 = *(const float4*)qp;
    float4 p1 = *(const float4*)(qp + 4);
    float q[8] = {p0.x, p0.y, p0.z, p0.w, p1.x, p1.y, p1.z, p1.w};
    const float* Qb = &Q8l[b * 64];
    const float* Lb = &L8l[b * 8];
    float s = 0.0f;
    #pragma unroll
    for (int i = 0; i < 8; ++i) {
      float inner = 2.0f * Lb[i];
      #pragma unroll
      for (int j = 0; j < 8; ++j) inner += Qb[i * 8 + j] * q[j];
      s += q[i] * inner;   // Σ_i q_i*(Σ_j Q_ij q_j + 2 L_i) = qᵀQq + 2L·q
    }
    s2a[b] += s;
  }
  #pragma unroll
  for (int bb = 0; bb < 2; ++bb) {
    float r = s2a[bb];
    r += __shfl_xor(r, 1);  r += __shfl_xor(r, 2);
    r += __shfl_xor(r, 4);  r += __shfl_xor(r, 8);
    r += __shfl_xor(r, 16);
    if ((tid & 31) == 0) atomicAdd(&s2L[bb], r);
  }
  __syncthreads();
  if (tid < 2) unsafeAtomicAdd(&acc[64 + tid], s2L[tid]);
}

// ---------------------------------------------------------------------------
// Kernel D: mean/rstd per batch from closed-form stats.
// ---------------------------------------------------------------------------
__global__ void kd_norm(float* __restrict__ acc)
{
  const int b = threadIdx.x;
  if (b < 2) {
    float s1 = 0.0f;
    for (int o = 0; o < 8; ++o) {
      float sa = 0.0f;
      for (int i = 0; i < 8; ++i)
        sa += acc[72 + b * 64 + o * 8 + i] * acc[48 + b * 8 + i];
      s1 += acc[344 + o] * sa;
    }
    s1 += (float)NVOX * acc[352];
    const float cnt = (float)DIMC * (float)NVOX;
    float mean = s1 / cnt;
    float e2 = (acc[64 + b] + (float)NVOX * acc[353]) / cnt;
    float var = e2 - mean * mean;
    acc[66 + b] = mean;
    acc[68 + b] = rsqrtf(var + 1e-5f);
  }
}

// ---------------------------------------------------------------------------
// Kernel E: out = GN( W_out @ (T q) + b ) via WMMA 16x16x4 (M-tiles of W_out).
// ---------------------------------------------------------------------------
__global__ __launch_bounds__(256) void ke_out(
    const float* __restrict__ qws, const float* __restrict__ Wout,
    const float* __restrict__ gamma, const float* __restrict__ beta,
    const float* __restrict__ acc, float* __restrict__ out)
{
  __shared__ __align__(16) float woL[1280];   // stride 10 per row (pairs aligned)
  __shared__ float Tl[128], gl[128], bl[128], mr[4];
  const int tid = threadIdx.x, lane = tid & 31, wv = tid >> 5;
  for (int i = tid; i < 1024; i += 256) woL[(i >> 3) * 10 + (i & 7)] = Wout[i];
  if (tid < 128) { Tl[tid] = acc[72 + tid]; gl[tid] = gamma[tid]; bl[tid] = beta[tid]; }
  if (tid < 4) mr[tid] = acc[66 + tid];       // mean[2], rstd[2]
  __syncthreads();

  const int m = lane & 15, half = lane >> 4;
  const int wavesTotal = gridDim.x * 8;
  for (int t = blockIdx.x * 8 + wv; t < (NBVOX / 16); t += wavesTotal) {
    const int vbase = t << 4;
    const int b  = vbase >> LOG2N;
    const int n0 = vbase & (NVOX - 1);
    const float* qp = qws + ((size_t)(vbase + m) << 3);
    float4 p0 = *(const float4*)qp;
    float4 p1 = *(const float4*)(qp + 4);
    float q[8] = {p0.x, p0.y, p0.z, p0.w, p1.x, p1.y, p1.z, p1.w};
    float a[8];
    const float* Tb = &Tl[b * 64];
    #pragma unroll
    for (int o = 0; o < 8; ++o) {
      float s = 0.0f;
      #pragma unroll
      for (int i = 0; i < 8; ++i) s += Tb[o * 8 + i] * q[i];
      a[o] = s;
    }
    v2f b0 = {a[2 * half],     a[2 * half + 1]};     // K=0..3 pairs
    v2f b1 = {a[4 + 2 * half], a[4 + 2 * half + 1]}; // K=4..7 pairs
    const float mean = mr[b], rstd = mr[2 + b];
    float* og = out + (size_t)b * DIMC * NVOX + n0 + m;
    #pragma unroll
    for (int mt = 0; mt < 8; ++mt) {
      v8f accv = {};
      v2f a0 = *(const v2f*)&woL[(mt * 16 + m) * 10 + 2 * half];
      accv = wmma4(a0, b0, accv);
      v2f a1 = *(const v2f*)&woL[(mt * 16 + m) * 10 + 4 + 2 * half];
      accv = wmma4(a1, b1, accv);
      #pragma unroll
      for (int j = 0; j < 8; ++j) {
        int c = mt * 16 + half * 8 + j;
        float y = (accv[j] - mean) * rstd * gl[c] + bl[c];
        og[(size_t)c * NVOX] = y;
      }
    }
  }
}

// ---------------------------------------------------------------------------
extern "C" void kernel_launch(void* const* d_in, const int* in_sizes, int n_in,
                              void* d_out, int out_size, void* d_ws, size_t ws_size,
                              hipStream_t stream) {
  const float* x     = (const float*)d_in[0];
  const float* query = (const float*)d_in[1];
  const float* Wqkv  = (const float*)d_in[2];
  const float* Wq    = (const float*)d_in[3];
  const float* Wout  = (const float*)d_in[4];
  const float* bout  = (const float*)d_in[5];
  const float* gamma = (const float*)d_in[6];
  const float* beta  = (const float*)d_in[7];
  float* out = (float*)d_out;

  float* qws = (float*)d_ws;                 // NBVOX*8 floats = 16 MiB
  float* acc = qws + (size_t)NBVOX * 8;      // small stats/context region

  (void)hipMemsetAsync(acc, 0, 72 * sizeof(float), stream);  // zero accumulators
  ka_project<<<512, 128, 0, stream>>>(x, query, Wqkv, Wq, qws, acc);
  kb_finalize<<<1, 256, 0, stream>>>(Wout, bout, acc);
  kc_stats<<<512, 256, 0, stream>>>(qws, acc);
  kd_norm<<<1, 64, 0, stream>>>(acc);
  ke_out<<<1024, 256, 0, stream>>>(qws, Wout, gamma, beta, acc, out);
}